// GNNlayer_68856915689819
// MI455X (gfx1250) — compile-verified
//
#include <hip/hip_runtime.h>
#include <hip/hip_bf16.h>

typedef float v2f __attribute__((ext_vector_type(2)));
typedef float v8f __attribute__((ext_vector_type(8)));

#define NB   4
#define CIN  270
#define COUT 256
#define NK   128
#define HW   65536
#define BN_EPS 1e-5f

// ---------------------------------------------------------------------------
// Kernel 1: y[b,d,p] = sum_c x[b,c,p] * w_ft[c,d]  via V_WMMA_F32_16X16X4_F32
//   M = output channel d, N = pixel p, K = input channel c.
//   Block = 256 thr (8 waves) covers ALL 256 channels x 64 pixels, so x is
//   streamed from HBM exactly once.  Each wave owns two 16-channel tiles
//   (d0 and d0+128) and shares every B(x) fragment across both A fragments:
//   per K-step 12 loads feed 8 WMMAs.  Main loop has an even trip count (66)
//   so the compiler's unroll-by-2 is exact (no mid-loop exit / acc copies);
//   channels 264..269 are handled by two peeled tail steps (270,271 masked).
//   Fused epilogue scatter-adds y into per-cluster channel sums.
// ---------------------------------------------------------------------------
__global__ __launch_bounds__(256) void k_gemm_y(
    const float* __restrict__ x, const float* __restrict__ w_ft,
    const int* __restrict__ index, float* __restrict__ y,
    float* __restrict__ sums)
{
  const int lane = threadIdx.x & 31;
  const int wave = threadIdx.x >> 5;
  const int half = lane >> 4;        // 0: lanes 0-15, 1: lanes 16-31
  const int l16  = lane & 15;
  const int b  = blockIdx.y;
  const int dl = wave * 16;          // low channel tile
  const int dh = 128 + wave * 16;    // high channel tile
  const int p0 = blockIdx.x * 64;

  __shared__ int s_idx[64];
  if (threadIdx.x < 64)
    s_idx[threadIdx.x] = index[(size_t)b * HW + p0 + threadIdx.x];
  __syncthreads();

  const float* xb = x + (size_t)b * CIN * HW;
  v8f accl[4] = {};
  v8f acch[4] = {};

  // A 16x4 f32 fragment layout: VGPR0 = K{0|2}, VGPR1 = K{1|3} per lane half.
  const int kk = half * 2;
#pragma unroll 2
  for (int k0 = 0; k0 < 264; k0 += 4) {
    const int c0 = k0 + kk;
    v2f al, ah;
    al.x = w_ft[(size_t)c0       * COUT + dl + l16];
    al.y = w_ft[(size_t)(c0 + 1) * COUT + dl + l16];
    ah.x = w_ft[(size_t)c0       * COUT + dh + l16];
    ah.y = w_ft[(size_t)(c0 + 1) * COUT + dh + l16];
    __builtin_prefetch(&xb[(size_t)(c0 + 8) * HW + p0 + lane], 0, 1);
#pragma unroll
    for (int j = 0; j < 4; ++j) {
      const int p = p0 + j * 16 + l16;
      v2f bf;
      bf.x = xb[(size_t)c0       * HW + p];
      bf.y = xb[(size_t)(c0 + 1) * HW + p];
      accl[j] = __builtin_amdgcn_wmma_f32_16x16x4_f32(
          false, al, false, bf, (short)0, accl[j], false, false);
      acch[j] = __builtin_amdgcn_wmma_f32_16x16x4_f32(
          false, ah, false, bf, (short)0, acch[j], false, false);
    }
  }
  { // peeled step: channels 264..267 (all valid)
    const int c0 = 264 + kk;
    v2f al, ah;
    al.x = w_ft[(size_t)c0       * COUT + dl + l16];
    al.y = w_ft[(size_t)(c0 + 1) * COUT + dl + l16];
    ah.x = w_ft[(size_t)c0       * COUT + dh + l16];
    ah.y = w_ft[(size_t)(c0 + 1) * COUT + dh + l16];
#pragma unroll
    for (int j = 0; j < 4; ++j) {
      const int p = p0 + j * 16 + l16;
      v2f bf;
      bf.x = xb[(size_t)c0       * HW + p];
      bf.y = xb[(size_t)(c0 + 1) * HW + p];
      accl[j] = __builtin_amdgcn_wmma_f32_16x16x4_f32(
          false, al, false, bf, (short)0, accl[j], false, false);
      acch[j] = __builtin_amdgcn_wmma_f32_16x16x4_f32(
          false, ah, false, bf, (short)0, acch[j], false, false);
    }
  }
  { // peeled masked step: 268,269 valid; 270,271 zero (branchless, EXEC full)
    const int c0 = 268 + kk;
    const float m0 = (c0     < CIN) ? 1.f : 0.f;
    const float m1 = (c0 + 1 < CIN) ? 1.f : 0.f;
    const int ca0 = (c0     < CIN) ? c0     : 0;
    const int ca1 = (c0 + 1 < CIN) ? c0 + 1 : 0;
    v2f al, ah;
    al.x = m0 * w_ft[(size_t)ca0 * COUT + dl + l16];
    al.y = m1 * w_ft[(size_t)ca1 * COUT + dl + l16];
    ah.x = m0 * w_ft[(size_t)ca0 * COUT + dh + l16];
    ah.y = m1 * w_ft[(size_t)ca1 * COUT + dh + l16];
#pragma unroll
    for (int j = 0; j < 4; ++j) {
      const int p = p0 + j * 16 + l16;
      v2f bf;
      bf.x = m0 * xb[(size_t)ca0 * HW + p];
      bf.y = m1 * xb[(size_t)ca1 * HW + p];
      accl[j] = __builtin_amdgcn_wmma_f32_16x16x4_f32(
          false, al, false, bf, (short)0, accl[j], false, false);
      acch[j] = __builtin_amdgcn_wmma_f32_16x16x4_f32(
          false, ah, false, bf, (short)0, acch[j], false, false);
    }
  }

  // C/D 16x16 f32 layout: lane l16 -> N(=pixel), vgpr r + half*8 -> M(=channel)
  float* yb = y    + (size_t)b * COUT * HW;
  float* sb = sums + (size_t)b * NK * COUT;
#pragma unroll
  for (int r = 0; r < 8; ++r) {
    const int d1 = dl + r + half * 8;
    const int d2 = dh + r + half * 8;
#pragma unroll
    for (int j = 0; j < 4; ++j) {
      const int pl = j * 16 + l16;
      const int kc = s_idx[pl];
      const float v1 = accl[j][r];
      const float v2 = acch[j][r];
      __builtin_nontemporal_store(v1, &yb[(size_t)d1 * HW + p0 + pl]);
      __builtin_nontemporal_store(v2, &yb[(size_t)d2 * HW + p0 + pl]);
      unsafeAtomicAdd(&sb[(size_t)kc * COUT + d1], v1);
      unsafeAtomicAdd(&sb[(size_t)kc * COUT + d2], v2);
    }
  }
}

// ---------------------------------------------------------------------------
// Kernel 2: per-(b,k) pixel counts via LDS histogram + global atomics
// ---------------------------------------------------------------------------
__global__ __launch_bounds__(256) void k_counts(const int* __restrict__ index,
                                                int* __restrict__ counts)
{
  __shared__ int s_hist[NK];
  const int b  = blockIdx.y;
  const int p0 = blockIdx.x * 4096;
  if (threadIdx.x < NK) s_hist[threadIdx.x] = 0;
  __syncthreads();
  for (int t = 0; t < 16; ++t)
    atomicAdd(&s_hist[index[(size_t)b * HW + p0 + t * 256 + threadIdx.x]], 1);
  __syncthreads();
  if (threadIdx.x < NK && s_hist[threadIdx.x] != 0)
    atomicAdd(&counts[b * NK + threadIdx.x], s_hist[threadIdx.x]);
}

// ---------------------------------------------------------------------------
// Kernel 3: Mm = w_adj @ w_adj^T   (256x256, K=256 — tiny)
// ---------------------------------------------------------------------------
__global__ __launch_bounds__(256) void k_mmat(const float* __restrict__ w_adj,
                                              float* __restrict__ Mm)
{
  const int i = blockIdx.x, j = threadIdx.x;
  __shared__ float s_wi[COUT];
  s_wi[j] = w_adj[(size_t)i * COUT + j];
  __syncthreads();
  float acc = 0.f;
  for (int d = 0; d < COUT; ++d) acc += s_wi[d] * w_adj[(size_t)j * COUT + d];
  Mm[(size_t)i * COUT + j] = acc;
}

// ---------------------------------------------------------------------------
// Kernel 4: rm[b,k,c] = sums[b,k,c] / max(counts[b,k], 1)
// ---------------------------------------------------------------------------
__global__ void k_rm(const float* __restrict__ sums, const int* __restrict__ counts,
                     float* __restrict__ rm)
{
  const int i = blockIdx.x * 256 + threadIdx.x;   // NB*NK*COUT = 131072
  const int bk = i / COUT;
  rm[i] = sums[i] / fmaxf((float)counts[bk], 1.f);
}

// ---------------------------------------------------------------------------
// Kernel 5: Pm[b,i,c] = sum_d rm[b,i,d]*Mm[d,c];  diag[b,i] = rm_i . M . rm_i
// ---------------------------------------------------------------------------
__global__ __launch_bounds__(256) void k_pm(const float* __restrict__ rm,
                                            const float* __restrict__ Mm,
                                            float* __restrict__ Pm,
                                            float* __restrict__ diag)
{
  const int i = blockIdx.x, b = blockIdx.y, c = threadIdx.x;
  __shared__ float s_r[COUT];
  s_r[c] = rm[((size_t)b * NK + i) * COUT + c];
  __syncthreads();
  float acc = 0.f;
  for (int d = 0; d < COUT; ++d) acc += s_r[d] * Mm[(size_t)d * COUT + c];
  Pm[((size_t)b * NK + i) * COUT + c] = acc;

  __shared__ float s_red[256];
  s_red[c] = acc * s_r[c];
  __syncthreads();
  for (int s = 128; s > 0; s >>= 1) {
    if (c < s) s_red[c] += s_red[c + s];
    __syncthreads();
  }
  if (c == 0) diag[b * NK + i] = s_red[0];
}

// ---------------------------------------------------------------------------
// Kernel 6: adj[b,i,j] = exp(-(diag_i + diag_j - 2*Gij)) + I   (M symmetric)
// ---------------------------------------------------------------------------
__global__ __launch_bounds__(128) void k_adj(const float* __restrict__ Pm,
                                             const float* __restrict__ rm,
                                             const float* __restrict__ diag,
                                             float* __restrict__ adj)
{
  const int i = blockIdx.x, b = blockIdx.y, j = threadIdx.x;
  const float* pi = Pm + ((size_t)b * NK + i) * COUT;
  const float* rj = rm + ((size_t)b * NK + j) * COUT;
  float g = 0.f;
  for (int c = 0; c < COUT; ++c) g += pi[c] * rj[c];
  const float quad = diag[b * NK + i] + diag[b * NK + j] - 2.f * g;
  adj[((size_t)b * NK + i) * NK + j] = expf(-quad) + (i == j ? 1.f : 0.f);
}

// ---------------------------------------------------------------------------
// Kernel 7: dinv[b,i] = rsqrt(sum_j adj[b,i,j])
// ---------------------------------------------------------------------------
__global__ void k_dinv(const float* __restrict__ adj, float* __restrict__ dinv)
{
  const int idx = blockIdx.x * 256 + threadIdx.x;   // NB*NK = 512
  const float* row = adj + (size_t)idx * NK;
  float s = 0.f;
  for (int j = 0; j < NK; ++j) s += row[j];
  dinv[idx] = rsqrtf(s);
}

// ---------------------------------------------------------------------------
// Kernel 8: am[b,i,c] = sum_j (adj_ij*dinv_i*dinv_j) * rm[b,j,c]
// ---------------------------------------------------------------------------
__global__ __launch_bounds__(256) void k_am(const float* __restrict__ adj,
                                            const float* __restrict__ dinv,
                                            const float* __restrict__ rm,
                                            float* __restrict__ am)
{
  const int i = blockIdx.x, b = blockIdx.y, c = threadIdx.x;
  __shared__ float s_w[NK];
  if (threadIdx.x < NK) {
    s_w[threadIdx.x] = adj[((size_t)b * NK + i) * NK + threadIdx.x] *
                       dinv[b * NK + i] * dinv[b * NK + threadIdx.x];
  }
  __syncthreads();
  float acc = 0.f;
  for (int j = 0; j < NK; ++j)
    acc += s_w[j] * rm[((size_t)b * NK + j) * COUT + c];
  am[((size_t)b * NK + i) * COUT + c] = acc;
}

// ---------------------------------------------------------------------------
// Kernel 9: in-place feat = relu(y + am[b, idx[p], d]) + BN statistics
//   one block per (d, pixel-chunk, b); am column cached in LDS
// ---------------------------------------------------------------------------
__global__ __launch_bounds__(256) void k_feat(float* __restrict__ y,
                                              const int* __restrict__ index,
                                              const float* __restrict__ am,
                                              float* __restrict__ chsum,
                                              float* __restrict__ chsumsq)
{
  const int d = blockIdx.x, b = blockIdx.z;
  const int p0 = blockIdx.y * 4096;
  __shared__ float s_am[NK];
  if (threadIdx.x < NK)
    s_am[threadIdx.x] = am[((size_t)b * NK + threadIdx.x) * COUT + d];
  __syncthreads();

  float* yp = y + ((size_t)b * COUT + d) * HW + p0;
  const int* ip = index + (size_t)b * HW + p0;
  float lsum = 0.f, lsq = 0.f;
#pragma unroll 4
  for (int t = 0; t < 16; ++t) {
    const int p = t * 256 + threadIdx.x;
    float v = __builtin_nontemporal_load(&yp[p]) + s_am[ip[p]];
    v = fmaxf(v, 0.f);
    __builtin_nontemporal_store(v, &yp[p]);
    lsum += v; lsq += v * v;
  }
  __shared__ float s_red[256];
  s_red[threadIdx.x] = lsum; __syncthreads();
  for (int s = 128; s > 0; s >>= 1) {
    if (threadIdx.x < s) s_red[threadIdx.x] += s_red[threadIdx.x + s];
    __syncthreads();
  }
  if (threadIdx.x == 0) unsafeAtomicAdd(&chsum[d], s_red[0]);
  __syncthreads();
  s_red[threadIdx.x] = lsq; __syncthreads();
  for (int s = 128; s > 0; s >>= 1) {
    if (threadIdx.x < s) s_red[threadIdx.x] += s_red[threadIdx.x + s];
    __syncthreads();
  }
  if (threadIdx.x == 0) unsafeAtomicAdd(&chsumsq[d], s_red[0]);
}

// ---------------------------------------------------------------------------
// Kernel 10: fold BN stats into per-channel scale/shift
// ---------------------------------------------------------------------------
__global__ void k_bnfin(const float* __restrict__ chsum, const float* __restrict__ chsumsq,
                        const float* __restrict__ gamma, const float* __restrict__ beta,
                        float* __restrict__ scale, float* __restrict__ shift)
{
  const int d = threadIdx.x;
  const float n = (float)NB * (float)HW;
  const float mean = chsum[d] / n;
  const float var  = chsumsq[d] / n - mean * mean;
  const float sc   = rsqrtf(var + BN_EPS) * gamma[d];
  scale[d] = sc;
  shift[d] = beta[d] - mean * sc;
}

// ---------------------------------------------------------------------------
// Kernel 11: in-place out = feat*scale[d] + shift[d]
// ---------------------------------------------------------------------------
__global__ __launch_bounds__(256) void k_bnapply(float* __restrict__ out,
                                                 const float* __restrict__ scale,
                                                 const float* __restrict__ shift)
{
  const size_t base = (size_t)blockIdx.x * 4096 + threadIdx.x;
#pragma unroll 4
  for (int t = 0; t < 16; ++t) {
    const size_t i = base + (size_t)t * 256;
    const int d = (int)((i >> 16) & (COUT - 1));   // HW = 2^16
    const float v = __builtin_nontemporal_load(&out[i]);
    __builtin_nontemporal_store(v * scale[d] + shift[d], &out[i]);
  }
}

// ---------------------------------------------------------------------------
extern "C" void kernel_launch(void* const* d_in, const int* in_sizes, int n_in,
                              void* d_out, int out_size, void* d_ws, size_t ws_size,
                              hipStream_t stream) {
  const float* x     = (const float*)d_in[0];
  const int*   index = (const int*)  d_in[1];
  const float* w_ft  = (const float*)d_in[2];
  const float* w_adj = (const float*)d_in[3];
  const float* gamma = (const float*)d_in[4];
  const float* beta  = (const float*)d_in[5];
  float* out = (float*)d_out;           // holds y, then feat, then BN output (in place)

  char* ws = (char*)d_ws;               // ~2.6 MB of small scratch
  float* sums   = (float*)(ws + 0);        // NB*NK*COUT f32   = 524288 B
  int*   counts = (int*)  (ws + 524288);   // NB*NK   i32      = 2048 B
  float* Mm     = (float*)(ws + 526336);   // COUT*COUT        = 262144 B
  float* rm     = (float*)(ws + 788480);   // NB*NK*COUT       = 524288 B
  float* Pm     = (float*)(ws + 1312768);  // NB*NK*COUT       = 524288 B
  float* diag   = (float*)(ws + 1837056);  // NB*NK            = 2048 B
  float* adj    = (float*)(ws + 1839104);  // NB*NK*NK         = 262144 B
  float* dinv   = (float*)(ws + 2101248);  // NB*NK            = 2048 B
  float* am     = (float*)(ws + 2103296);  // NB*NK*COUT       = 524288 B
  float* chsum  = (float*)(ws + 2627584);  // COUT             = 1024 B
  float* chsq   = (float*)(ws + 2628608);  // COUT             = 1024 B
  float* scale  = (float*)(ws + 2629632);  // COUT             = 1024 B
  float* shift  = (float*)(ws + 2630656);  // COUT             = 1024 B

  hipMemsetAsync(ws, 0, 526336, stream);             // sums + counts
  hipMemsetAsync(ws + 2627584, 0, 2048, stream);     // chsum + chsumsq

  k_gemm_y<<<dim3(1024, NB), 256, 0, stream>>>(x, w_ft, index, out, sums);
  k_counts<<<dim3(16, NB), 256, 0, stream>>>(index, counts);
  k_mmat<<<COUT, COUT, 0, stream>>>(w_adj, Mm);
  k_rm<<<512, 256, 0, stream>>>(sums, counts, rm);
  k_pm<<<dim3(NK, NB), 256, 0, stream>>>(rm, Mm, Pm, diag);
  k_adj<<<dim3(NK, NB), NK, 0, stream>>>(Pm, rm, diag, adj);
  k_dinv<<<2, 256, 0, stream>>>(adj, dinv);
  k_am<<<dim3(NK, NB), 256, 0, stream>>>(adj, dinv, rm, am);
  k_feat<<<dim3(COUT, 16, NB), 256, 0, stream>>>(out, index, am, chsum, chsq);
  k_bnfin<<<1, 256, 0, stream>>>(chsum, chsq, gamma, beta, scale, shift);
  k_bnapply<<<16384, 256, 0, stream>>>(out, scale, shift);
}